// DifferentiableProjectionLayer_23888608100662
// MI455X (gfx1250) — compile-verified
//
#include <hip/hip_runtime.h>
#include <hip/hip_bf16.h>

#define W_CONST 0.01f
#define EPSF    1e-8f
#define FLT_INF __builtin_inff()
#define PPB     8          // points per block in the ray kernel

typedef __attribute__((ext_vector_type(2))) float v2f;
typedef __attribute__((ext_vector_type(8))) float v8f;

// ---------------------------------------------------------------------------
// Kernel A: precompute vaug = (v, |v|^2) and triangle (tri0, e1, e2) as float4
// ---------------------------------------------------------------------------
__global__ void precompute_kernel(const float* __restrict__ verts,
                                  const int*   __restrict__ faces,
                                  float4* __restrict__ vaug,
                                  float4* __restrict__ t0,
                                  float4* __restrict__ e1,
                                  float4* __restrict__ e2,
                                  int V, int F)
{
    int i = blockIdx.x * blockDim.x + threadIdx.x;
    if (i < V) {
        float x = verts[3*i+0], y = verts[3*i+1], z = verts[3*i+2];
        vaug[i] = make_float4(x, y, z, x*x + y*y + z*z);
    }
    if (i < F) {
        int a = faces[3*i+0], b = faces[3*i+1], c = faces[3*i+2];
        float ax = verts[3*a+0], ay = verts[3*a+1], az = verts[3*a+2];
        float bx = verts[3*b+0], by = verts[3*b+1], bz = verts[3*b+2];
        float cx = verts[3*c+0], cy = verts[3*c+1], cz = verts[3*c+2];
        t0[i] = make_float4(ax, ay, az, 0.f);
        e1[i] = make_float4(bx-ax, by-ay, bz-az, 0.f);
        e2[i] = make_float4(cx-ax, cy-ay, cz-az, 0.f);
    }
}

// Branchless sorted insert into 8-register top-k (ascending: d0 smallest).
#define TOP8_INSERT(KEY, IDX) do {                                         \
    float _k = (KEY); int _x = (IDX);                                      \
    if (_k < d7) {                                                         \
        bool c0=_k<d0,c1=_k<d1,c2=_k<d2,c3=_k<d3,c4=_k<d4,c5=_k<d5,c6=_k<d6;\
        d7 = c6?d6:_k;         i7 = c6?i6:_x;                              \
        d6 = c5?d5:(c6?_k:d6); i6 = c5?i5:(c6?_x:i6);                      \
        d5 = c4?d4:(c5?_k:d5); i5 = c4?i4:(c5?_x:i5);                      \
        d4 = c3?d3:(c4?_k:d4); i4 = c3?i3:(c4?_x:i4);                      \
        d3 = c2?d2:(c3?_k:d3); i3 = c2?i2:(c3?_x:i3);                      \
        d2 = c1?d1:(c2?_k:d2); i2 = c1?i1:(c2?_x:i2);                      \
        d1 = c0?d0:(c1?_k:d1); i1 = c0?i0:(c1?_x:i1);                      \
        d0 = c0?_k:d0;         i0 = c0?_x:i0;                              \
    }                                                                      \
} while (0)

// ---------------------------------------------------------------------------
// Kernel B: KNN (K=8) + coarse normal.
// One wave handles 16 points. Distance keys via V_WMMA_F32_16X16X4_F32:
//   key[m,n] = |v_m|^2 - 2 x_n . v_m  ==  d2[m,n] - |x_n|^2  (same ordering)
// A[m,k] = (vx,vy,vz,|v|^2), B[k,n] = (-2x,-2y,-2z,1), C = 0.
// A operand is loaded as one aligned b64 per lane (lane-dependent address,
// wave-coalesced to 256B) and software-pipelined one 16-vertex tile ahead.
// Each lane keeps top-8 of its 8-row half; halves merge with __shfl_xor(16).
// ---------------------------------------------------------------------------
__global__ void knn_kernel(const float*  __restrict__ x,
                           const float*  __restrict__ vnorm,
                           const float4* __restrict__ vaug,
                           float4* __restrict__ ncbuf,
                           float4* __restrict__ v1buf,
                           int N, int V)
{
    const int lane  = threadIdx.x & 31;
    const int wave  = threadIdx.x >> 5;
    const int half  = lane >> 4;       // 0: K=0,1 / rows 0-7; 1: K=2,3 / rows 8-15
    const int l16   = lane & 15;
    const int ptile = (blockIdx.x * (blockDim.x >> 5) + wave) << 4;
    if (ptile + 16 > N) return;        // wave-uniform guard (EXEC stays all-ones)
    const int p = ptile + l16;

    const float px = x[3*p+0], py = x[3*p+1], pz = x[3*p+2];
    v2f bmat;
    bmat.x = half ? (-2.0f * pz) : (-2.0f * px);   // K=2 : K=0
    bmat.y = half ? 1.0f         : (-2.0f * py);   // K=3 : K=1

    float d0=FLT_INF,d1=FLT_INF,d2=FLT_INF,d3=FLT_INF,
          d4=FLT_INF,d5=FLT_INF,d6=FLT_INF,d7=FLT_INF;
    int   i0=0,i1=0,i2=0,i3=0,i4=0,i5=0,i6=0,i7=0;

    const v2f* __restrict__ vaug2 = (const v2f*)vaug;  // 2 K-pairs per vertex
    const int aoff = l16*2 + half;     // this lane's K-pair within a tile
    v2f anext = vaug2[aoff];           // tile vt=0, pipelined load

    for (int vt = 0; vt < V; vt += 16) {
        v2f amat = anext;
        int nvt = vt + 16; if (nvt > V - 16) nvt = V - 16;   // clamp (no branch)
        anext = vaug2[nvt*2 + aoff];                          // prefetch next tile
        __builtin_prefetch((const char*)&vaug2[(nvt+16)*2 + aoff], 0, 0);

        v8f c = {};
        c = __builtin_amdgcn_wmma_f32_16x16x4_f32(
                /*neg_a=*/false, amat, /*neg_b=*/false, bmat,
                /*c_mod=*/(short)0, c, /*reuse_a=*/false, /*reuse_b=*/false);
        const int mb = vt + (half << 3);
        TOP8_INSERT(c[0], mb+0);  TOP8_INSERT(c[1], mb+1);
        TOP8_INSERT(c[2], mb+2);  TOP8_INSERT(c[3], mb+3);
        TOP8_INSERT(c[4], mb+4);  TOP8_INSERT(c[5], mb+5);
        TOP8_INSERT(c[6], mb+6);  TOP8_INSERT(c[7], mb+7);
    }

    // Merge the two half-lane lists for the same point (snapshot first!).
    float q0=__shfl_xor(d0,16,32), q1=__shfl_xor(d1,16,32), q2=__shfl_xor(d2,16,32),
          q3=__shfl_xor(d3,16,32), q4=__shfl_xor(d4,16,32), q5=__shfl_xor(d5,16,32),
          q6=__shfl_xor(d6,16,32), q7=__shfl_xor(d7,16,32);
    int   j0=__shfl_xor(i0,16,32), j1=__shfl_xor(i1,16,32), j2=__shfl_xor(i2,16,32),
          j3=__shfl_xor(i3,16,32), j4=__shfl_xor(i4,16,32), j5=__shfl_xor(i5,16,32),
          j6=__shfl_xor(i6,16,32), j7=__shfl_xor(i7,16,32);
    TOP8_INSERT(q0,j0); TOP8_INSERT(q1,j1); TOP8_INSERT(q2,j2); TOP8_INSERT(q3,j3);
    TOP8_INSERT(q4,j4); TOP8_INSERT(q5,j5); TOP8_INSERT(q6,j6); TOP8_INSERT(q7,j7);

    if (half == 0) {
        // Exact d2 for the 8 winners (reference uses true d2 for the weights).
        float sw = 0.f, tx = 0.f, ty = 0.f, tz = 0.f;
#define ACC(II) { float4 vv = vaug[(II)];                                  \
        float ddx=px-vv.x, ddy=py-vv.y, ddz=pz-vv.z;                       \
        float dd = ddx*ddx + ddy*ddy + ddz*ddz;                            \
        float inv = 1.0f / fmaxf(dd, EPSF);                                \
        tx += vnorm[3*(II)+0]*inv; ty += vnorm[3*(II)+1]*inv;              \
        tz += vnorm[3*(II)+2]*inv; sw += inv; }
        ACC(i0); ACC(i1); ACC(i2); ACC(i3); ACC(i4); ACC(i5); ACC(i6); ACC(i7);
#undef ACC
        float4 v1 = vaug[i0];                       // list sorted: i0 = nearest
        float vx = px - v1.x, vy = py - v1.y, vz = pz - v1.z;
        float dv1 = fmaxf(vx*vx + vy*vy + vz*vz, EPSF);
        float invdir = 1.0f / (W_CONST * dv1);
        tx += vx * invdir; ty += vy * invdir; tz += vz * invdir;
        float W = sw + (1.0f / W_CONST);
        tx /= W; ty /= W; tz /= W;
        float len = sqrtf(tx*tx + ty*ty + tz*tz) + 1e-8f;
        ncbuf[p] = make_float4(tx/len, ty/len, tz/len, 0.f);
        v1buf[p] = make_float4(v1.x, v1.y, v1.z, 0.f);
    }
}

// ---------------------------------------------------------------------------
// Kernel C: two ray-mesh first-hit sweeps + combine.
// PPB=8 points per 256-thread block: each 48B triangle load feeds 16 ray
// tests (8x less L2 traffic than 1 point/block). Per point, tv / qvec / E2.q
// are shared between its two rays. Reduction: wave32 shfl-xor min trees,
// then an 8-wave x 16-value LDS merge; threads 0..7 finalize one point each.
// ---------------------------------------------------------------------------
__global__ void ray_kernel(const float*  __restrict__ x,
                           const float4* __restrict__ t0,
                           const float4* __restrict__ e1v,
                           const float4* __restrict__ e2v,
                           const float4* __restrict__ ncbuf,
                           const float4* __restrict__ v1buf,
                           float* __restrict__ out, int N, int F)
{
    __shared__ float red[8][2*PPB];
    const int tid   = threadIdx.x;
    const int lane  = tid & 31;
    const int wid   = tid >> 5;
    const int pbase = blockIdx.x * PPB;

    float ox[PPB], oy[PPB], oz[PPB];
    float ax1[PPB], ay1[PPB], az1[PPB];   // dir1 = -nc
    float ax2[PPB], ay2[PPB], az2[PPB];   // dir2 = normalize(v1 - x)
    float t1[PPB], t2[PPB];
#pragma unroll
    for (int k = 0; k < PPB; ++k) {
        const int p = pbase + k;
        ox[k] = x[3*p+0]; oy[k] = x[3*p+1]; oz[k] = x[3*p+2];
        float4 nc = ncbuf[p];
        float4 v1 = v1buf[p];
        ax1[k] = -nc.x; ay1[k] = -nc.y; az1[k] = -nc.z;
        float fx = v1.x - ox[k], fy = v1.y - oy[k], fz = v1.z - oz[k];
        float fl = sqrtf(fx*fx + fy*fy + fz*fz) + 1e-8f;
        ax2[k] = fx/fl; ay2[k] = fy/fl; az2[k] = fz/fl;
        t1[k] = FLT_INF; t2[k] = FLT_INF;
    }

    for (int f = tid; f < F; f += blockDim.x) {
        __builtin_prefetch((const char*)&t0[f + blockDim.x], 0, 0);
        float4 a  = t0[f];
        float4 E1 = e1v[f];
        float4 E2 = e2v[f];
#pragma unroll
        for (int k = 0; k < PPB; ++k) {
            // per-(point,tri) terms shared by both rays
            float tvx = ox[k] - a.x, tvy = oy[k] - a.y, tvz = oz[k] - a.z;
            float qx = tvy*E1.z - tvz*E1.y;
            float qy = tvz*E1.x - tvx*E1.z;
            float qz = tvx*E1.y - tvy*E1.x;
            float e2q = E2.x*qx + E2.y*qy + E2.z*qz;
            // ray 1
            {
                float dx = ax1[k], dy = ay1[k], dz = az1[k];
                float pxv = dy*E2.z - dz*E2.y;
                float pyv = dz*E2.x - dx*E2.z;
                float pzv = dx*E2.y - dy*E2.x;
                float det = E1.x*pxv + E1.y*pyv + E1.z*pzv;
                bool ok = fabsf(det) > EPSF;
                float inv = ok ? (1.0f/det) : 0.0f;
                float u = (tvx*pxv + tvy*pyv + tvz*pzv) * inv;
                float v = (dx*qx + dy*qy + dz*qz) * inv;
                float t = e2q * inv;
                bool valid = ok && (u >= 0.f) && (v >= 0.f) && ((u+v) <= 1.f) && (t > EPSF);
                t1[k] = fminf(t1[k], valid ? t : FLT_INF);
            }
            // ray 2
            {
                float dx = ax2[k], dy = ay2[k], dz = az2[k];
                float pxv = dy*E2.z - dz*E2.y;
                float pyv = dz*E2.x - dx*E2.z;
                float pzv = dx*E2.y - dy*E2.x;
                float det = E1.x*pxv + E1.y*pyv + E1.z*pzv;
                bool ok = fabsf(det) > EPSF;
                float inv = ok ? (1.0f/det) : 0.0f;
                float u = (tvx*pxv + tvy*pyv + tvz*pzv) * inv;
                float v = (dx*qx + dy*qy + dz*qz) * inv;
                float t = e2q * inv;
                bool valid = ok && (u >= 0.f) && (v >= 0.f) && ((u+v) <= 1.f) && (t > EPSF);
                t2[k] = fminf(t2[k], valid ? t : FLT_INF);
            }
        }
    }

    // wave32 min-reduction, then cross-wave LDS merge
#pragma unroll
    for (int k = 0; k < PPB; ++k) {
        float a = t1[k], b = t2[k];
        for (int m = 16; m >= 1; m >>= 1) {
            a = fminf(a, __shfl_xor(a, m, 32));
            b = fminf(b, __shfl_xor(b, m, 32));
        }
        t1[k] = a; t2[k] = b;
    }
    if (lane == 0) {
#pragma unroll
        for (int k = 0; k < PPB; ++k) {
            red[wid][2*k+0] = t1[k];
            red[wid][2*k+1] = t2[k];
        }
    }
    __syncthreads();

    if (tid < PPB) {
        const int k = tid, p = pbase + k;
        float T1 = FLT_INF, T2 = FLT_INF;
#pragma unroll
        for (int w = 0; w < 8; ++w) {
            T1 = fminf(T1, red[w][2*k+0]);
            T2 = fminf(T2, red[w][2*k+1]);
        }
        float Ox = x[3*p+0], Oy = x[3*p+1], Oz = x[3*p+2];
        float4 nc = ncbuf[p];
        float4 v1 = v1buf[p];
        float D1x = -nc.x, D1y = -nc.y, D1z = -nc.z;
        float fx = v1.x - Ox, fy = v1.y - Oy, fz = v1.z - Oz;
        float fl = sqrtf(fx*fx + fy*fy + fz*fz) + 1e-8f;
        float D2x = fx/fl, D2y = fy/fl, D2z = fz/fl;
        bool h1 = (T1 < FLT_INF), h2 = (T2 < FLT_INF);
        float s1 = h1 ? T1 : 0.f, s2 = h2 ? T2 : 0.f;
        float l1x = Ox + D1x*s1, l1y = Oy + D1y*s1, l1z = Oz + D1z*s1;
        float l2x = Ox + D2x*s2, l2y = Oy + D2y*s2, l2z = Oz + D2z*s2;
        float xcx = h1 ? l1x : (h2 ? l2x : v1.x);
        float xcy = h1 ? l1y : (h2 ? l2y : v1.y);
        float xcz = h1 ? l1z : (h2 ? l2z : v1.z);
        float s = (Ox - xcx)*nc.x + (Oy - xcy)*nc.y + (Oz - xcz)*nc.z;
        out[3*p+0] = xcx; out[3*p+1] = xcy; out[3*p+2] = xcz;   // xc
        out[3*N + p] = s;                                       // s
        out[4*N + 3*p+0] = nc.x;                                // nc
        out[4*N + 3*p+1] = nc.y;
        out[4*N + 3*p+2] = nc.z;
    }
}

// ---------------------------------------------------------------------------
extern "C" void kernel_launch(void* const* d_in, const int* in_sizes, int n_in,
                              void* d_out, int out_size, void* d_ws, size_t ws_size,
                              hipStream_t stream)
{
    const float* x     = (const float*)d_in[0];
    const float* verts = (const float*)d_in[1];
    const float* vnorm = (const float*)d_in[2];
    const int*   faces = (const int*)d_in[3];
    const int N = in_sizes[0] / 3;   // 2048
    const int V = in_sizes[1] / 3;   // 8192
    const int F = in_sizes[3] / 3;   // 8192

    char* ws = (char*)d_ws;
    size_t off = 0;
    float4* vaug = (float4*)(ws + off); off += (size_t)V * sizeof(float4);
    float4* t0   = (float4*)(ws + off); off += (size_t)F * sizeof(float4);
    float4* e1   = (float4*)(ws + off); off += (size_t)F * sizeof(float4);
    float4* e2   = (float4*)(ws + off); off += (size_t)F * sizeof(float4);
    float4* ncb  = (float4*)(ws + off); off += (size_t)N * sizeof(float4);
    float4* v1b  = (float4*)(ws + off); off += (size_t)N * sizeof(float4);
    (void)ws_size; (void)n_in; (void)out_size;

    const int M = (V > F) ? V : F;
    precompute_kernel<<<(M + 255) / 256, 256, 0, stream>>>(
        verts, faces, vaug, t0, e1, e2, V, F);

    // 128 threads = 4 waves = 64 points per block
    knn_kernel<<<(N + 63) / 64, 128, 0, stream>>>(
        x, vnorm, vaug, ncb, v1b, N, V);

    // 8 points per 256-thread block
    ray_kernel<<<(N + PPB - 1) / PPB, 256, 0, stream>>>(
        x, t0, e1, e2, ncb, v1b, (float*)d_out, N, F);
}